// PlasticLayer_1168231105117
// MI455X (gfx1250) — compile-verified
//
#include <hip/hip_runtime.h>
#include <math.h>

typedef __attribute__((ext_vector_type(2))) float v2f;
typedef __attribute__((ext_vector_type(4))) float v4f;
typedef __attribute__((ext_vector_type(8))) float v8f;

#define E_DIM 4096
#define B_DIM 8
#define PF_ROWS 32   // prefetch distance in K-rows

// -----------------------------------------------------------------------------
// Phase 1: yout = tanh(yin @ (w + alpha*hebb) + input)
// One wave (32 lanes) per 16-column output tile. K reduced in steps of 4 with
// V_WMMA_F32_16X16X4_F32. B-operand (M = w + alpha*hebb) is formed in
// registers so each of w/alpha/hebb is streamed from HBM exactly once.
//
// f32 WMMA register layouts (CDNA5 ISA 7.12.2):
//   A (16x4):  lanes 0-15 hold rows 0-15, VGPR0=K0, VGPR1=K1;
//              lanes 16-31 hold rows 0-15, VGPR0=K2, VGPR1=K3.
//   B (4x16):  VGPRv, lane l: row = 2*v + (l>=16), col = l%16.
//   C (16x16): VGPRr, lane l: row = r + 8*(l>=16), col = l%16.
// -----------------------------------------------------------------------------
template <bool PF>
__device__ __forceinline__ v8f k_step(
    int k, const float* __restrict__ yrow, float amask,
    const float* __restrict__ w, const float* __restrict__ alpha,
    const float* __restrict__ hebb, int half, int col, v8f acc)
{
    // ---- A tile: yin[0:16, k:k+4] (rows >= 8 zeroed, branchless) ----
    const int ka = k + 2 * half;
    v2f a;
    a.x = yrow[ka + 0] * amask;
    a.y = yrow[ka + 1] * amask;

    // ---- B tile: M[k:k+4, j0:j0+16], M = w + alpha*hebb ----
    const int i0 = (k + half) * E_DIM + col;
    const int i1 = i0 + 2 * E_DIM;
    v2f b;
    b.x = fmaf(alpha[i0], hebb[i0], w[i0]);
    b.y = fmaf(alpha[i1], hebb[i1], w[i1]);

    if (PF) {
        // Immediate-offset prefetch of the three parameter streams ahead.
        __builtin_prefetch(w     + i0 + PF_ROWS * E_DIM, 0, 1);
        __builtin_prefetch(alpha + i0 + PF_ROWS * E_DIM, 0, 1);
        __builtin_prefetch(hebb  + i0 + PF_ROWS * E_DIM, 0, 1);
    }

    // D = A x B + C  (16x16x4 f32)
    return __builtin_amdgcn_wmma_f32_16x16x4_f32(
        /*neg_a=*/false, a, /*neg_b=*/false, b,
        /*c_mod=*/(short)0, acc, /*reuse_a=*/false, /*reuse_b=*/false);
}

__global__ __launch_bounds__(32) void plastic_yout_wmma(
    const float* __restrict__ input,   // [B, E]
    const float* __restrict__ yin,     // [B, E]
    const float* __restrict__ hebb,    // [E, E]
    const float* __restrict__ w,       // [E, E]
    const float* __restrict__ alpha,   // [E, E]
    float* __restrict__ yout)          // [B, E]
{
    const int j0   = blockIdx.x * 16;      // first output column of this tile
    const int lane = threadIdx.x;          // 0..31
    const int half = lane >> 4;            // 0: lanes 0-15, 1: lanes 16-31
    const int l16  = lane & 15;

    // A-operand: row of yin this lane feeds (rows 8..15 are zero padding).
    // Branchless masking keeps EXEC all-1s, as WMMA requires.
    const float  amask = (l16 < B_DIM) ? 1.0f : 0.0f;
    const float* yrow  = yin + (l16 & (B_DIM - 1)) * E_DIM;

    const int col = j0 + l16;

    v8f acc = {};
    int k = 0;

    // Main loop: unconditional prefetch (no branches in steady state).
    #pragma unroll 4
    for (; k < E_DIM - PF_ROWS; k += 4)
        acc = k_step<true>(k, yrow, amask, w, alpha, hebb, half, col, acc);

    // Tail: last PF_ROWS K-rows, no prefetch.
    #pragma unroll 4
    for (; k < E_DIM; k += 4)
        acc = k_step<false>(k, yrow, amask, w, alpha, hebb, half, col, acc);

    // Epilogue: rows 0..7 live in VGPRs 0..7 of lanes 0..15 (half==0).
    if (half == 0) {
        #pragma unroll
        for (int r = 0; r < B_DIM; ++r) {
            const int o = r * E_DIM + col;
            yout[o] = tanhf(acc[r] + input[o]);
        }
    }
}

// -----------------------------------------------------------------------------
// Phase 2: hebb' = (1-eta)*hebb + eta * outer(yin[0], yout[0])
// Pure streaming elementwise pass, float4 vectorized (row stride E_DIM is a
// multiple of 4, so 16B alignment holds).
// -----------------------------------------------------------------------------
__global__ __launch_bounds__(256) void hebb_update(
    const float* __restrict__ hebb,     // [E, E]
    const float* __restrict__ yin,      // [B, E] (row 0 used)
    const float* __restrict__ yout,     // [B, E] (row 0 used)
    const float* __restrict__ eta_p,    // [1]
    float* __restrict__ hebb_out)       // [E, E]
{
    const float eta = eta_p[0];
    const float om  = 1.0f - eta;

    const long idx  = (long)blockIdx.x * blockDim.x + threadIdx.x;
    const long base = idx * 4;                      // 4 consecutive elements
    if (base >= (long)E_DIM * E_DIM) return;

    const int row = (int)(base >> 12);              // base / 4096
    const int c0  = (int)(base & (E_DIM - 1));      // base % 4096

    const float yi = eta * yin[row];                // eta * yin[0][row]
    const v4f h  = *(const v4f*)(hebb + base);
    const v4f yo = *(const v4f*)(yout + c0);        // yout[0][c0..c0+3]

    v4f o;
    o.x = om * h.x + yi * yo.x;
    o.y = om * h.y + yi * yo.y;
    o.z = om * h.z + yi * yo.z;
    o.w = om * h.w + yi * yo.w;
    *(v4f*)(hebb_out + base) = o;
}

extern "C" void kernel_launch(void* const* d_in, const int* in_sizes, int n_in,
                              void* d_out, int out_size, void* d_ws, size_t ws_size,
                              hipStream_t stream) {
    const float* input = (const float*)d_in[0];   // [8, 4096]
    const float* yin   = (const float*)d_in[1];   // [8, 4096]
    const float* hebb  = (const float*)d_in[2];   // [4096, 4096]
    const float* w     = (const float*)d_in[3];   // [4096, 4096]
    const float* alpha = (const float*)d_in[4];   // [4096, 4096]
    const float* eta   = (const float*)d_in[5];   // [1]

    float* yout     = (float*)d_out;              // first output: [8, 4096]
    float* hebb_new = yout + B_DIM * E_DIM;       // second output: [4096, 4096]

    // Phase 1: 256 column tiles, one wave32 each.
    plastic_yout_wmma<<<E_DIM / 16, 32, 0, stream>>>(input, yin, hebb, w, alpha, yout);

    // Phase 2: stream-ordered after phase 1; reads yout[0] from d_out.
    const long n4 = (long)E_DIM * E_DIM / 4;      // float4 elements
    hebb_update<<<(int)((n4 + 255) / 256), 256, 0, stream>>>(hebb, yin, yout, eta, hebb_new);
}